// NeighborhoodReasoner_70257075028111
// MI455X (gfx1250) — compile-verified
//
#include <hip/hip_runtime.h>
#include <hip/hip_bf16.h>

typedef float v2f __attribute__((ext_vector_type(2)));
typedef float v8f __attribute__((ext_vector_type(8)));

#define D     128
#define KNN   16
#define NB    20000     // nodes per batch (tiles never cross batches: 20000 % 32 == 0)
#define TM    32        // rows (nodes) per wave tile = 2 WMMA M-subtiles
#define WAVES 4
#define BLOCK (WAVES * 32)

// One K=128 GEMM pass over a 32-row tile (two 16-row M-subtiles sharing each B fragment):
//   acc0/acc1[nt] += A(sub-tile rows from LDS) x W(128x128 row-major, L2-resident)
// A fragment layout (V_WMMA_F32_16X16X4_F32): lanes 0-15 hold M=lane, K={k0,k0+1};
// lanes 16-31 hold M=lane-16, K={k0+2,k0+3}. B mirrored: VGPR0 row k0(+2), VGPR1 row k0+1(+3).
__device__ __forceinline__ void gemm_pass2(const float* __restrict__ arow0,
                                           const float* __restrict__ arow1,
                                           const float* __restrict__ W,
                                           int koff, int nl,
                                           v8f acc0[8], v8f acc1[8]) {
    #pragma unroll 4
    for (int kc = 0; kc < 32; ++kc) {
        const int k0 = kc * 4 + koff;
        v2f a0 = *(const v2f*)(arow0 + k0);
        v2f a1 = *(const v2f*)(arow1 + k0);
        #pragma unroll
        for (int nt = 0; nt < 8; ++nt) {
            v2f bb;
            bb.x = W[(k0 + 0) * D + nt * 16 + nl];
            bb.y = W[(k0 + 1) * D + nt * 16 + nl];
            acc0[nt] = __builtin_amdgcn_wmma_f32_16x16x4_f32(
                false, a0, false, bb, (short)0, acc0[nt], false, false);
            acc1[nt] = __builtin_amdgcn_wmma_f32_16x16x4_f32(
                false, a1, false, bb, (short)0, acc1[nt], false, false);
        }
    }
}

// Same, but A = (x - mean) computed on the fly from LDS rows.
__device__ __forceinline__ void gemm_pass2_diff(const float* __restrict__ xrow0,
                                                const float* __restrict__ mrow0,
                                                const float* __restrict__ xrow1,
                                                const float* __restrict__ mrow1,
                                                const float* __restrict__ W,
                                                int koff, int nl,
                                                v8f acc0[8], v8f acc1[8]) {
    #pragma unroll 4
    for (int kc = 0; kc < 32; ++kc) {
        const int k0 = kc * 4 + koff;
        v2f x0 = *(const v2f*)(xrow0 + k0);
        v2f m0 = *(const v2f*)(mrow0 + k0);
        v2f x1 = *(const v2f*)(xrow1 + k0);
        v2f m1 = *(const v2f*)(mrow1 + k0);
        v2f a0;  a0.x = x0.x - m0.x;  a0.y = x0.y - m0.y;
        v2f a1;  a1.x = x1.x - m1.x;  a1.y = x1.y - m1.y;
        #pragma unroll
        for (int nt = 0; nt < 8; ++nt) {
            v2f bb;
            bb.x = W[(k0 + 0) * D + nt * 16 + nl];
            bb.y = W[(k0 + 1) * D + nt * 16 + nl];
            acc0[nt] = __builtin_amdgcn_wmma_f32_16x16x4_f32(
                false, a0, false, bb, (short)0, acc0[nt], false, false);
            acc1[nt] = __builtin_amdgcn_wmma_f32_16x16x4_f32(
                false, a1, false, bb, (short)0, acc1[nt], false, false);
        }
    }
}

__global__ __launch_bounds__(BLOCK) void NeighborhoodReasoner_kernel(
    const float* __restrict__ emb, const int* __restrict__ knn,
    const float* __restrict__ W1, const float* __restrict__ b1,
    const float* __restrict__ W2, const float* __restrict__ b2,
    float* __restrict__ out, int total_nodes)
{
    extern __shared__ float smem[];   // per wave: x tile (32x128) + mean/h tile (32x128)
    const int tid  = threadIdx.x;
    const int wave = tid >> 5;
    const int lane = tid & 31;
    const int tile = blockIdx.x * WAVES + wave;
    const int node0 = tile * TM;
    if (node0 >= total_nodes) return;           // uniform per wave; no block barriers used

    float* xbuf = smem + wave * (2 * TM * D);
    float* mbuf = xbuf + TM * D;

    const bool hi   = lane >= 16;
    const int  nl   = lane & 15;
    const int  koff = hi ? 2 : 0;

    // ---------------- Phase 1: gather neighbors -> mean; stage x and mean in LDS --------
    {
        const int d0 = lane * 4;                 // each lane owns 4 contiguous dims
        for (int m = 0; m < TM; ++m) {
            const int  g     = node0 + m;
            const int  b     = g / NB;
            const long bbase = (long)b * NB * D;
            int myidx = 0;
            if (lane < KNN) myidx = knn[(long)g * KNN + lane];
            float4 acc = make_float4(0.f, 0.f, 0.f, 0.f);
            #pragma unroll
            for (int j = 0; j < KNN; ++j) {
                const int idx = __shfl(myidx, j, 32);
                const float4 v = *(const float4*)(emb + bbase + (long)idx * D + d0);
                acc.x += v.x; acc.y += v.y; acc.z += v.z; acc.w += v.w;
            }
            const float s = 1.0f / (float)KNN;
            float4 mv = make_float4(acc.x * s, acc.y * s, acc.z * s, acc.w * s);
            *(float4*)(mbuf + m * D + d0) = mv;
            *(float4*)(xbuf + m * D + d0) = *(const float4*)(emb + (long)g * D + d0);
        }
    }

    const float* xrow0 = xbuf + nl * D;
    const float* xrow1 = xbuf + (16 + nl) * D;
    const float* mrow0 = mbuf + nl * D;
    const float* mrow1 = mbuf + (16 + nl) * D;

    // ---------------- Phase 2: GEMM1  h = gelu(x@W1a + mean@W1b + (x-mean)@W1c + b1) ----
    v8f acc0[8], acc1[8];
    #pragma unroll
    for (int nt = 0; nt < 8; ++nt) {
        const float bv = b1[nt * 16 + nl];       // C/D layout: col = nt*16 + (lane&15), all 8 rows
        acc0[nt] = (v8f){bv, bv, bv, bv, bv, bv, bv, bv};
        acc1[nt] = acc0[nt];
    }
    gemm_pass2     (xrow0, xrow1,               W1 + 0 * D * D, koff, nl, acc0, acc1); // x part
    gemm_pass2     (mrow0, mrow1,               W1 + 1 * D * D, koff, nl, acc0, acc1); // mean part
    gemm_pass2_diff(xrow0, mrow0, xrow1, mrow1, W1 + 2 * D * D, koff, nl, acc0, acc1); // x-mean part

    // exact GELU; stage h into LDS (reuse mean buffer) for GEMM2's A operand
    #pragma unroll
    for (int nt = 0; nt < 8; ++nt) {
        #pragma unroll
        for (int gg = 0; gg < 8; ++gg) {
            const int row = gg + (hi ? 8 : 0);   // C/D layout: VGPR gg -> M = gg (+8 for hi half)
            const float v0 = acc0[nt][gg];
            const float v1 = acc1[nt][gg];
            mbuf[row * D        + nt * 16 + nl] = 0.5f * v0 * (1.0f + erff(v0 * 0.70710678118654752f));
            mbuf[(row + 16) * D + nt * 16 + nl] = 0.5f * v1 * (1.0f + erff(v1 * 0.70710678118654752f));
        }
    }

    // ---------------- Phase 3: GEMM2  update = h@W2 + b2 --------------------------------
    #pragma unroll
    for (int nt = 0; nt < 8; ++nt) {
        const float bv = b2[nt * 16 + nl];
        acc0[nt] = (v8f){bv, bv, bv, bv, bv, bv, bv, bv};
        acc1[nt] = acc0[nt];
    }
    gemm_pass2(mrow0, mrow1, W2, koff, nl, acc0, acc1);

    // ---------------- Phase 4: residual (x still staged in LDS) + coalesced store -------
    #pragma unroll
    for (int nt = 0; nt < 8; ++nt) {
        #pragma unroll
        for (int gg = 0; gg < 8; ++gg) {
            const int row = gg + (hi ? 8 : 0);
            xbuf[row * D        + nt * 16 + nl] += acc0[nt][gg];
            xbuf[(row + 16) * D + nt * 16 + nl] += acc1[nt][gg];
        }
    }
    {
        const int d0 = lane * 4;
        for (int m = 0; m < TM; ++m) {
            const float4 v = *(const float4*)(xbuf + m * D + d0);
            *(float4*)(out + (long)(node0 + m) * D + d0) = v;
        }
    }
}

extern "C" void kernel_launch(void* const* d_in, const int* in_sizes, int n_in,
                              void* d_out, int out_size, void* d_ws, size_t ws_size,
                              hipStream_t stream) {
    const float* emb = (const float*)d_in[0];   // [B,N,D] f32
    const int*   knn = (const int*)  d_in[1];   // [B,N,K] i32
    const float* W1  = (const float*)d_in[2];   // [3D,D]  f32
    const float* b1  = (const float*)d_in[3];   // [D]
    const float* W2  = (const float*)d_in[4];   // [D,D]
    const float* b2  = (const float*)d_in[5];   // [D]
    float* out = (float*)d_out;

    const int total_nodes = in_sizes[0] / D;    // B*N = 80000
    const int tiles  = (total_nodes + TM - 1) / TM;
    const int blocks = (tiles + WAVES - 1) / WAVES;
    const size_t shmem = (size_t)WAVES * 2 * TM * D * sizeof(float);  // 128 KB / block

    NeighborhoodReasoner_kernel<<<blocks, BLOCK, shmem, stream>>>(
        emb, knn, W1, b1, W2, b2, out, total_nodes);
}